// DKVMN_13357348290825
// MI455X (gfx1250) — compile-verified
//
#include <hip/hip_runtime.h>
#include <hip/hip_bf16.h>
#include <math.h>

typedef __attribute__((ext_vector_type(2))) float v2f;
typedef __attribute__((ext_vector_type(8))) float v8f;

#define BB   128
#define SS   200
#define DKK  128
#define DVV  64
#define NUMQ 10000
#define ROWS (BB*SS)      // 25600
#define MTILES (ROWS/16)  // 1600

__device__ __forceinline__ v8f wmma_f32(v2f a, v2f b, v8f c) {
    // V_WMMA_F32_16X16X4_F32  (8-arg pattern: neg_a, A, neg_b, B, c_mod, C, reuse_a, reuse_b)
    return __builtin_amdgcn_wmma_f32_16x16x4_f32(false, a, false, b, (short)0, c, false, false);
}

// Fast reciprocal: single V_RCP_F32 (TRANS32) instead of IEEE div expansion.
__device__ __forceinline__ float fast_rcp_(float x) { return __builtin_amdgcn_rcpf(x); }

__device__ __forceinline__ float sigmoidf_(float x) {
    return fast_rcp_(1.0f + __expf(-x));           // v_exp_f32 + v_add + v_rcp_f32
}

// gfx1250 native V_TANH_F32 (TRANS32); branchless exp-based fallback.
__device__ __forceinline__ float fast_tanhf_(float x) {
#if defined(__has_builtin)
#if __has_builtin(__builtin_amdgcn_tanhf)
    return __builtin_amdgcn_tanhf(x);
#else
    return 1.0f - 2.0f * fast_rcp_(__expf(2.0f * x) + 1.0f);
#endif
#else
    return 1.0f - 2.0f * fast_rcp_(__expf(2.0f * x) + 1.0f);
#endif
}

// -------------------------------------------------------------------------
// Phase 1: per 16-row M-tile compute
//   logits = k_emb[q] . Mk^T  -> softmax -> w   [rows x 64]
//   e = sigmoid(v_emb[x] . eW + eb)             [rows x 128]
//   a = tanh   (v_emb[x] . aW + ab)             [rows x 128]
// Gathers are folded into the WMMA A-fragment loads.
// 128 threads = 4 waves; 20 N-tiles (4 w + 8 e + 8 a), 5 per wave.
// -------------------------------------------------------------------------
__global__ __launch_bounds__(128) void dkvmn_phase1(
    const int*   __restrict__ qseq, const int* __restrict__ cseq,
    const float* __restrict__ k_emb, const float* __restrict__ v_emb,
    const float* __restrict__ Mk,
    const float* __restrict__ eW, const float* __restrict__ eb,
    const float* __restrict__ aW, const float* __restrict__ ab,
    float* __restrict__ w_ws, float* __restrict__ e_ws, float* __restrict__ a_ws)
{
    __shared__ float logits[16][DVV];

    const int tid   = threadIdx.x;
    const int wave  = tid >> 5;
    const int lane  = tid & 31;
    const int lrow  = lane & 15;
    const int lhalf = lane >> 4;

    const int mbase = blockIdx.x * 16;
    const int row   = mbase + lrow;
    const int q     = qseq[row];
    const int c     = cseq[row];
    const float* arow_k = k_emb + (size_t)q * DKK;
    const float* arow_v = v_emb + ((size_t)q + (size_t)NUMQ * (size_t)c) * DKK;

    for (int t = wave; t < 20; t += 4) {
        int kind, nbase;                      // 0 = w (B = Mk^T), 1 = e, 2 = a
        if (t < 4)       { kind = 0; nbase = t * 16; }
        else if (t < 12) { kind = 1; nbase = (t - 4) * 16; }
        else             { kind = 2; nbase = (t - 12) * 16; }
        const float* arow = (kind == 0) ? arow_k : arow_v;
        const int ncol = nbase + lrow;

        v8f acc = {};
        if (kind == 0) {
            const float* brow = Mk + (size_t)ncol * DKK;   // B[k][n] = Mk[n][k]
            for (int k0 = 0; k0 < DKK; k0 += 4) {
                const int kk = k0 + 2 * lhalf;
                v2f a; a.x = arow[kk]; a.y = arow[kk + 1];
                v2f b; b.x = brow[kk]; b.y = brow[kk + 1];
                acc = wmma_f32(a, b, acc);
            }
            #pragma unroll
            for (int i = 0; i < 8; ++i)
                logits[i + 8 * lhalf][ncol] = acc[i];
        } else {
            const float* W    = (kind == 1) ? eW : aW;
            const float  bias = (kind == 1) ? eb[ncol] : ab[ncol];
            for (int k0 = 0; k0 < DKK; k0 += 4) {
                const int kk = k0 + 2 * lhalf;
                v2f a; a.x = arow[kk]; a.y = arow[kk + 1];
                v2f b;
                b.x = W[(size_t)kk       * DKK + ncol];
                b.y = W[(size_t)(kk + 1) * DKK + ncol];
                acc = wmma_f32(a, b, acc);
            }
            // Arm-separated activation/store loops: one block per activation
            // kind instead of a per-element predicated select.
            if (kind == 1) {
                #pragma unroll
                for (int i = 0; i < 8; ++i)
                    e_ws[(size_t)(mbase + i + 8 * lhalf) * DKK + ncol] =
                        sigmoidf_(acc[i] + bias);
            } else {
                #pragma unroll
                for (int i = 0; i < 8; ++i)
                    a_ws[(size_t)(mbase + i + 8 * lhalf) * DKK + ncol] =
                        fast_tanhf_(acc[i] + bias);
            }
        }
    }
    __syncthreads();

    // softmax over the 64 memory slots, one row per thread (three LDS passes)
    if (tid < 16) {
        float m = -1e30f;
        #pragma unroll
        for (int v = 0; v < DVV; ++v) m = fmaxf(m, logits[tid][v]);
        float s = 0.f;
        #pragma unroll
        for (int v = 0; v < DVV; ++v) s += __expf(logits[tid][v] - m);
        const float inv = fast_rcp_(s);
        float* dst = w_ws + (size_t)(mbase + tid) * DVV;
        #pragma unroll
        for (int v = 0; v < DVV; ++v) dst[v] = __expf(logits[tid][v] - m) * inv;
    }
}

// -------------------------------------------------------------------------
// Phase 2: sequential memory scan, one block per batch element.
// Mvt (64x128 f32 = 32KB) lives in LDS. 512 threads: k = tid&127,
// v-quarter g = tid>>7. Each cell (v,k) is owned by exactly one thread,
// so read-before-update + rank-1 update need no intra-step barrier.
// -------------------------------------------------------------------------
__global__ __launch_bounds__(512) void dkvmn_scan(
    const float* __restrict__ Mv0,
    const float* __restrict__ w_ws, const float* __restrict__ e_ws,
    const float* __restrict__ a_ws, float* __restrict__ reads_ws)
{
    __shared__ float M[DVV][DKK];      // 32 KB
    __shared__ float wv[DVV];
    __shared__ float part[4][DKK];

    const int tid = threadIdx.x;
    const int k   = tid & (DKK - 1);
    const int g   = tid >> 7;          // 0..3
    const int b   = blockIdx.x;

    for (int i = tid; i < DVV * DKK; i += 512)
        (&M[0][0])[i] = Mv0[i];
    __syncthreads();

    for (int s = 0; s < SS; ++s) {
        const size_t base = (size_t)b * SS + s;
        if (tid < DVV) wv[tid] = w_ws[base * DVV + tid];
        const float ek = e_ws[base * DKK + k];
        const float ak = a_ws[base * DKK + k];
        __syncthreads();                       // wv visible; prev part reads done

        float acc = 0.f;
        const int v0 = g * 16;
        #pragma unroll
        for (int j = 0; j < 16; ++j) {
            const int v = v0 + j;
            const float wgt = wv[v];
            const float m   = M[v][k];
            acc = fmaf(wgt, m, acc);                       // read BEFORE update
            M[v][k] = m * (1.0f - wgt * ek) + wgt * ak;    // erase + add
        }
        part[g][k] = acc;
        __syncthreads();                       // partials visible

        if (g == 0)
            reads_ws[base * DKK + k] = part[0][k] + part[1][k] + part[2][k] + part[3][k];
    }
}

// -------------------------------------------------------------------------
// Phase 3: f = tanh([reads | k] . fW + fb), p = sigmoid(f . pW + pb).
// One block per 16-row M-tile; 8 waves cover the 8 N-tiles; K=256 (reads
// half from ws, k-half gathered from k_emb on the fly). Final dot with pW
// reduced through LDS.
// -------------------------------------------------------------------------
__global__ __launch_bounds__(256) void dkvmn_phase3(
    const int*   __restrict__ qseq, const float* __restrict__ k_emb,
    const float* __restrict__ reads_ws,
    const float* __restrict__ fW, const float* __restrict__ fb,
    const float* __restrict__ pW, const float* __restrict__ pb,
    float* __restrict__ out)
{
    __shared__ float fT[16][DKK];      // activated f tile (8 KB)
    __shared__ float pp[16][16];

    const int tid   = threadIdx.x;
    const int wave  = tid >> 5;        // 0..7 -> N-tile
    const int lane  = tid & 31;
    const int lrow  = lane & 15;
    const int lhalf = lane >> 4;

    const int mbase = blockIdx.x * 16;
    const int row   = mbase + lrow;
    const float* arow_r = reads_ws + (size_t)row * DKK;
    const float* arow_k = k_emb + (size_t)qseq[row] * DKK;

    const int ncol = wave * 16 + lrow;

    v8f acc = {};
    for (int k0 = 0; k0 < 2 * DKK; k0 += 4) {      // k0 % 4 == 0: chunk never straddles 128
        const int kk = k0 + 2 * lhalf;
        v2f a;
        if (k0 < DKK) { a.x = arow_r[kk];       a.y = arow_r[kk + 1]; }
        else          { a.x = arow_k[kk - DKK]; a.y = arow_k[kk + 1 - DKK]; }
        v2f b;
        b.x = fW[(size_t)kk       * DKK + ncol];
        b.y = fW[(size_t)(kk + 1) * DKK + ncol];
        acc = wmma_f32(a, b, acc);
    }
    const float bias = fb[ncol];
    #pragma unroll
    for (int i = 0; i < 8; ++i)
        fT[i + 8 * lhalf][ncol] = fast_tanhf_(acc[i] + bias);
    __syncthreads();

    {   // p-partial: 16 threads per row, strided over the 128 cols
        const int r = tid >> 4;
        const int j = tid & 15;
        float partial = 0.f;
        #pragma unroll
        for (int c = j; c < DKK; c += 16)
            partial = fmaf(fT[r][c], pW[c], partial);
        pp[r][j] = partial;
    }
    __syncthreads();
    if (tid < 16) {
        float s = pb[0];
        #pragma unroll
        for (int j = 0; j < 16; ++j) s += pp[tid][j];
        out[mbase + tid] = sigmoidf_(s);
    }
}

// -------------------------------------------------------------------------
extern "C" void kernel_launch(void* const* d_in, const int* in_sizes, int n_in,
                              void* d_out, int out_size, void* d_ws, size_t ws_size,
                              hipStream_t stream) {
    const int*   qseq  = (const int*)  d_in[0];
    const int*   cseq  = (const int*)  d_in[1];
    const float* k_emb = (const float*)d_in[2];
    const float* v_emb = (const float*)d_in[3];
    const float* Mk    = (const float*)d_in[4];
    const float* Mv0   = (const float*)d_in[5];
    const float* fW    = (const float*)d_in[6];
    const float* fb    = (const float*)d_in[7];
    const float* eW    = (const float*)d_in[8];
    const float* eb    = (const float*)d_in[9];
    const float* aW    = (const float*)d_in[10];
    const float* ab    = (const float*)d_in[11];
    const float* pW    = (const float*)d_in[12];
    const float* pb    = (const float*)d_in[13];
    float* out = (float*)d_out;

    // workspace layout (floats)
    float* ws      = (float*)d_ws;
    float* w_ws    = ws;                                   // 25600*64
    float* e_ws    = w_ws + (size_t)ROWS * DVV;            // 25600*128
    float* a_ws    = e_ws + (size_t)ROWS * DKK;            // 25600*128
    float* reads_ws= a_ws + (size_t)ROWS * DKK;            // 25600*128

    dkvmn_phase1<<<MTILES, 128, 0, stream>>>(qseq, cseq, k_emb, v_emb, Mk,
                                             eW, eb, aW, ab, w_ws, e_ws, a_ws);
    dkvmn_scan  <<<BB,     512, 0, stream>>>(Mv0, w_ws, e_ws, a_ws, reads_ws);
    dkvmn_phase3<<<MTILES, 256, 0, stream>>>(qseq, k_emb, reads_ws,
                                             fW, fb, pW, pb, out);
}